// im_choose_46351287059051
// MI455X (gfx1250) — compile-verified
//
#include <hip/hip_runtime.h>
#include <hip/hip_bf16.h>

#define BATCH 8
#define CIN   256
#define NPTS  4096
#define OUTC  256
#define TOPK  1024

typedef float v2f __attribute__((ext_vector_type(2)));
typedef float v8f __attribute__((ext_vector_type(8)));

// ---------------------------------------------------------------------------
// Kernel 1: v[b,ci] = sum_c (Wq @ l1[:, :, 0])[b,c] * Wk[c,ci]
// Two GEMMs M=256, N=16 (8 batches + zero pad), K=256 via V_WMMA_F32_16X16X4_F32.
// 16 waves, one 16-row M-tile per wave.
// A layout (16x4 f32): lane<16 -> M=lane, K={0,1}; lane>=16 -> M=lane-16, K={2,3}
// B layout (4x16 f32): lane<16 -> N=lane, K={0,1}; lane>=16 -> N=lane-16, K={2,3}
// D layout (16x16 f32): VGPR r: lane<16 -> D[r][lane]; lane>=16 -> D[8+r][lane-16]
// ---------------------------------------------------------------------------
__global__ void __launch_bounds__(512) qkv_wmma_kernel(
    const float* __restrict__ l1, const float* __restrict__ Wq,
    const float* __restrict__ Wk, float* __restrict__ v_out)
{
    __shared__ float L0[CIN][16];   // L0[ci][b], cols 8..15 zero
    __shared__ float Q0[OUTC][16];  // Q0[c][b]

    const int tid   = threadIdx.x;
    const int wave  = tid >> 5;          // 0..15 : M-tile
    const int lane  = tid & 31;
    const int mrow  = lane & 15;         // M (or N) within tile
    const int khalf = (lane >> 4) * 2;   // 0 or 2 : K sub-pair
    const int mbase = wave * 16;

    for (int i = tid; i < CIN * 16; i += 512) {
        int ci = i >> 4, b = i & 15;
        L0[ci][b] = (b < BATCH) ? l1[((size_t)b * CIN + ci) * NPTS] : 0.0f;
    }
    __syncthreads();

    // GEMM 1: Q0 = Wq(256x256) x L0(256x16)
    v8f acc = {0.f,0.f,0.f,0.f,0.f,0.f,0.f,0.f};
    for (int k0 = 0; k0 < CIN; k0 += 4) {
        v2f a, bm;
        a.x  = Wq[(size_t)(mbase + mrow) * CIN + k0 + khalf];
        a.y  = Wq[(size_t)(mbase + mrow) * CIN + k0 + khalf + 1];
        bm.x = L0[k0 + khalf][mrow];
        bm.y = L0[k0 + khalf + 1][mrow];
        acc = __builtin_amdgcn_wmma_f32_16x16x4_f32(
                  false, a, false, bm, (short)0, acc, false, false);
    }
    {
        int mo = mbase + ((lane >> 4) ? 8 : 0);
        #pragma unroll
        for (int r = 0; r < 8; ++r) Q0[mo + r][mrow] = acc[r];
    }
    __syncthreads();

    // GEMM 2: Vt(256x16) = Wk^T(256x256) x Q0(256x16); A[m][k] = Wk[k*CIN + m]
    v8f acc2 = {0.f,0.f,0.f,0.f,0.f,0.f,0.f,0.f};
    for (int k0 = 0; k0 < OUTC; k0 += 4) {
        v2f a, bm;
        a.x  = Wk[(size_t)(k0 + khalf) * CIN + (mbase + mrow)];
        a.y  = Wk[(size_t)(k0 + khalf + 1) * CIN + (mbase + mrow)];
        bm.x = Q0[k0 + khalf][mrow];
        bm.y = Q0[k0 + khalf + 1][mrow];
        acc2 = __builtin_amdgcn_wmma_f32_16x16x4_f32(
                   false, a, false, bm, (short)0, acc2, false, false);
    }
    {
        int mo = mbase + ((lane >> 4) ? 8 : 0);
        if (mrow < BATCH) {
            #pragma unroll
            for (int r = 0; r < 8; ++r)
                v_out[(size_t)mrow * CIN + mo + r] = acc2[r];
        }
    }
}

// ---------------------------------------------------------------------------
// Kernel 2: energy0[b,m] = v[b,:] . l1[b,:,m], emitted as sortable u64 key.
// key = (~monotone_u32(energy)) << 32 | m  => ascending sort == descending
// value with lower-index-first tie break (matches lax.top_k semantics).
// ---------------------------------------------------------------------------
__global__ void __launch_bounds__(256) energy_kernel(
    const float* __restrict__ l1, const float* __restrict__ v,
    unsigned long long* __restrict__ keys)
{
    __shared__ float vs[CIN];
    const int b = blockIdx.y;
    const int m = blockIdx.x * 256 + threadIdx.x;
    vs[threadIdx.x] = v[(size_t)b * CIN + threadIdx.x];
    __syncthreads();

    const float* base = l1 + (size_t)b * CIN * NPTS + m;
    float e = 0.0f;
    #pragma unroll 8
    for (int ci = 0; ci < CIN; ++ci)
        e = fmaf(vs[ci], base[(size_t)ci * NPTS], e);

    unsigned int bits = __float_as_uint(e);
    unsigned int s = (bits & 0x80000000u) ? ~bits : (bits | 0x80000000u);
    keys[(size_t)b * NPTS + m] =
        ((unsigned long long)(~s) << 32) | (unsigned int)m;
}

// ---------------------------------------------------------------------------
// Kernel 3: per-batch full bitonic sort of 4096 u64 keys in LDS (32 KB);
// first TOPK entries give topk indices in jax.lax.top_k order.
// ---------------------------------------------------------------------------
__global__ void __launch_bounds__(1024) topk_kernel(
    const unsigned long long* __restrict__ keys_in,
    unsigned int* __restrict__ topk)
{
    __shared__ unsigned long long sk[NPTS];
    const int b = blockIdx.x;
    const int tid = threadIdx.x;

    for (int i = tid; i < NPTS; i += 1024)
        sk[i] = keys_in[(size_t)b * NPTS + i];
    __syncthreads();

    for (int k = 2; k <= NPTS; k <<= 1) {
        for (int j = k >> 1; j > 0; j >>= 1) {
            #pragma unroll
            for (int t0 = 0; t0 < NPTS; t0 += 1024) {
                int t = t0 + tid;
                int ixj = t ^ j;
                if (ixj > t) {
                    unsigned long long a = sk[t], c = sk[ixj];
                    bool up = ((t & k) == 0);
                    if ((a > c) == up) { sk[t] = c; sk[ixj] = a; }
                }
            }
            __syncthreads();
        }
    }
    if (tid < TOPK)
        topk[(size_t)b * TOPK + tid] = (unsigned int)(sk[tid] & 0xFFFFFFFFu);
}

// ---------------------------------------------------------------------------
// Kernel 4: gather l1 columns (coalesced 256-wide writes; reads hit L2 since
// l1 just streamed through the 192MB L2) + xyz1 columns.
// out = [ l1_new (B,TOPK,CIN) | p1_new (B,TOPK,3) ] flat.
// ---------------------------------------------------------------------------
__global__ void __launch_bounds__(256) gather_kernel(
    const float* __restrict__ l1, const float* __restrict__ xyz,
    const unsigned int* __restrict__ topk, float* __restrict__ out)
{
    const int j = blockIdx.x;   // 0..TOPK-1
    const int b = blockIdx.y;   // 0..BATCH-1
    const int c = threadIdx.x;  // 0..CIN-1
    const unsigned int idx = topk[(size_t)b * TOPK + j];

    out[((size_t)b * TOPK + j) * CIN + c] =
        l1[((size_t)b * CIN + c) * NPTS + idx];

    if (c < 3) {
        float* p1 = out + (size_t)BATCH * TOPK * CIN;
        p1[((size_t)b * TOPK + j) * 3 + c] =
            xyz[((size_t)b * 3 + c) * NPTS + idx];
    }
}

extern "C" void kernel_launch(void* const* d_in, const int* in_sizes, int n_in,
                              void* d_out, int out_size, void* d_ws, size_t ws_size,
                              hipStream_t stream)
{
    (void)in_sizes; (void)n_in; (void)out_size; (void)ws_size;
    const float* l1   = (const float*)d_in[0];
    const float* xyz1 = (const float*)d_in[1];
    const float* Wq   = (const float*)d_in[2];
    const float* Wk   = (const float*)d_in[3];
    // d_in[4] = top_k scalar (1024), compile-time constant here.

    char* ws = (char*)d_ws;
    float* v = (float*)ws;                                      //  8 KB
    unsigned long long* keys =
        (unsigned long long*)(ws + BATCH * CIN * sizeof(float)); // 256 KB
    unsigned int* topk =
        (unsigned int*)(ws + BATCH * CIN * sizeof(float)
                           + (size_t)BATCH * NPTS * sizeof(unsigned long long)); // 32 KB

    qkv_wmma_kernel<<<1, 512, 0, stream>>>(l1, Wq, Wk, v);
    energy_kernel<<<dim3(NPTS / 256, BATCH), 256, 0, stream>>>(l1, v, keys);
    topk_kernel<<<BATCH, 1024, 0, stream>>>(keys, topk);
    gather_kernel<<<dim3(TOPK, BATCH), 256, 0, stream>>>(l1, xyz1, topk, (float*)d_out);
}